// BipartiteResMRConv_73796128079943
// MI455X (gfx1250) — compile-verified
//
#include <hip/hip_runtime.h>
#include <cstdint>

// Problem constants (match reference)
#define NUM_SRC   100000
#define NUM_DST   100000
#define NUM_EDGES 800000
#define WIDTH     128
#define LEAKY     0.01f
#define LDS_STRIDE 132           // 128 + 4 pad: bank = (4*m + k) % 64, conflict-free
#define KEY_SENTINEL (-2147483647 - 1)

typedef __attribute__((ext_vector_type(2))) float v2f;
typedef __attribute__((ext_vector_type(8))) float v8f;

// Monotone float <-> int key mapping (total order matches float order).
// encode never produces INT_MIN for non-NaN input -> INT_MIN is a safe
// "empty segment" sentinel.
__device__ __forceinline__ int f2key(float f) {
    int i = __float_as_int(f);
    return i >= 0 ? i : (i ^ 0x7fffffff);
}
__device__ __forceinline__ float key2f_or_zero(int k) {
    if (k == KEY_SENTINEL) return 0.0f;                 // empty segment -> 0
    return __int_as_float(k >= 0 ? k : (k ^ 0x7fffffff));
}

// ---------------- Kernel 1: init max-key buffer ----------------
__global__ void init_keys_kernel(int* __restrict__ keys) {
    int g = blockIdx.x * blockDim.x + threadIdx.x;      // n/4 threads
    ((int4*)keys)[g] = make_int4(KEY_SENTINEL, KEY_SENTINEL, KEY_SENTINEL, KEY_SENTINEL);
}

// ---------------- Kernel 2: per-edge scatter max ----------------
// 32 threads per edge, 4 features per thread (float4 gathers, 4 int atomics).
__global__ void edge_scatter_max_kernel(const float* __restrict__ x_src,
                                        const float* __restrict__ x_dst,
                                        const int*   __restrict__ e,
                                        int*         __restrict__ keys) {
    int g    = blockIdx.x * blockDim.x + threadIdx.x;
    int edge = g >> 5;
    int f    = (g & 31) << 2;
    int s = e[edge];
    int d = e[NUM_EDGES + edge];
    float4 vd = *(const float4*)(x_dst + d * WIDTH + f);
    float4 vs = *(const float4*)(x_src + s * WIDTH + f);
    int* kp = keys + d * WIDTH + f;
    atomicMax(kp + 0, f2key(vd.x - vs.x));   // -> global_atomic_max_i32
    atomicMax(kp + 1, f2key(vd.y - vs.y));
    atomicMax(kp + 2, f2key(vd.z - vs.z));
    atomicMax(kp + 3, f2key(vd.w - vs.w));
}

// ---------------- Kernel 3: WMMA GEMM + bias + LeakyReLU + residual ----------------
// Block: 256 threads = 8 waves. Block covers 16 nodes x 128 output cols;
// wave w computes the 16x16 tile at columns [16w, 16w+16).
__global__ __launch_bounds__(256)
void gemm_wmma_kernel(const float* __restrict__ x_dst,
                      const int*   __restrict__ keys,
                      const float* __restrict__ W,     // [256,128] row-major
                      const float* __restrict__ b,     // [128]
                      float*       __restrict__ out) {
    __shared__ float lhs[16 * LDS_STRIDE];

    const int tid   = threadIdx.x;
    const int lane  = tid & 31;
    const int wave  = tid >> 5;
    const int kh    = lane >> 4;      // K-half select for A/B frags (0 or 1)
    const int ml    = lane & 15;      // A: row M; B/C/D: col N within tile
    const int node0 = blockIdx.x * 16;
    const int jcol  = wave * 16 + ml; // absolute output column for this lane

    v8f c = {};

    // ---- Phase 1: LHS = x_dst tile, W rows [0,128) ----
    {
        int row  = tid >> 4;              // 0..15
        int col0 = (tid & 15) * 8;        // 0..120
        const float4* src = (const float4*)(x_dst + (node0 + row) * WIDTH + col0);
        float4 p0 = src[0], p1 = src[1];
        float* dp = &lhs[row * LDS_STRIDE + col0];
        *(float4*)(dp)     = p0;
        *(float4*)(dp + 4) = p1;
    }
    __syncthreads();

    const float* Wp = W + jcol;
    #pragma unroll
    for (int kk = 0; kk < 128; kk += 4) {
        // A frag: 32-bit A 16x4 layout -> lane holds (M=ml, K=kk+2*kh, kk+2*kh+1)
        float2 av = *(const float2*)(&lhs[ml * LDS_STRIDE + kk + 2 * kh]);
        v2f a = {av.x, av.y};
        // B frag: rows K=kk+2*kh, +1, col jcol
        int kb = kk + 2 * kh;
        v2f bf = {Wp[kb * WIDTH], Wp[(kb + 1) * WIDTH]};
        c = __builtin_amdgcn_wmma_f32_16x16x4_f32(false, a, false, bf,
                                                  (short)0, c, false, false);
    }
    __syncthreads();

    // ---- Phase 2: LHS = segment maxes (decoded), W rows [128,256) ----
    {
        int row  = tid >> 4;
        int col0 = (tid & 15) * 8;
        const int* src = keys + (node0 + row) * WIDTH + col0;
        int4 k0 = *(const int4*)(src);
        int4 k1 = *(const int4*)(src + 4);
        float* dp = &lhs[row * LDS_STRIDE + col0];
        dp[0] = key2f_or_zero(k0.x); dp[1] = key2f_or_zero(k0.y);
        dp[2] = key2f_or_zero(k0.z); dp[3] = key2f_or_zero(k0.w);
        dp[4] = key2f_or_zero(k1.x); dp[5] = key2f_or_zero(k1.y);
        dp[6] = key2f_or_zero(k1.z); dp[7] = key2f_or_zero(k1.w);
    }
    __syncthreads();

    const float* Wp2 = W + 128 * WIDTH + jcol;
    #pragma unroll
    for (int kk = 0; kk < 128; kk += 4) {
        float2 av = *(const float2*)(&lhs[ml * LDS_STRIDE + kk + 2 * kh]);
        v2f a = {av.x, av.y};
        int kb = kk + 2 * kh;
        v2f bf = {Wp2[kb * WIDTH], Wp2[(kb + 1) * WIDTH]};
        c = __builtin_amdgcn_wmma_f32_16x16x4_f32(false, a, false, bf,
                                                  (short)0, c, false, false);
    }

    // ---- Epilogue: bias + LeakyReLU + residual ----
    // C/D layout: VGPR i -> row M = i (lanes 0-15) or i+8 (lanes 16-31), col = ml.
    float bj = b[jcol];
    #pragma unroll
    for (int i = 0; i < 8; ++i) {
        int m = i + 8 * kh;
        int off = (node0 + m) * WIDTH + jcol;
        float h = c[i] + bj;
        h = h > 0.0f ? h : LEAKY * h;
        out[off] = x_dst[off] + h;
    }
}

extern "C" void kernel_launch(void* const* d_in, const int* in_sizes, int n_in,
                              void* d_out, int out_size, void* d_ws, size_t ws_size,
                              hipStream_t stream) {
    const float* x_src = (const float*)d_in[0];
    const float* x_dst = (const float*)d_in[1];
    const int*   e     = (const int*)d_in[2];   // [2, NUM_EDGES] int32
    const float* W     = (const float*)d_in[3]; // [256,128]
    const float* b     = (const float*)d_in[4]; // [128]
    float* out = (float*)d_out;
    int*   keys = (int*)d_ws;                   // NUM_DST*WIDTH ints = 51.2 MB

    // 1) init sentinel keys: 12.8M ints / 4 per thread / 256 per block
    init_keys_kernel<<<(NUM_DST * WIDTH / 4) / 256, 256, 0, stream>>>(keys);
    // 2) edge scatter-max: 800000 edges * 32 threads / 256 per block
    edge_scatter_max_kernel<<<(NUM_EDGES * 32) / 256, 256, 0, stream>>>(x_src, x_dst, e, keys);
    // 3) GEMM + epilogue: 16 nodes per block
    gemm_wmma_kernel<<<NUM_DST / 16, 256, 0, stream>>>(x_dst, keys, W, b, out);
}